// SAdapter_609885356267
// MI455X (gfx1250) — compile-verified
//
#include <hip/hip_runtime.h>
#include <hip/hip_bf16.h>

typedef __attribute__((ext_vector_type(2))) float v2f;
typedef __attribute__((ext_vector_type(8))) float v8f;

#define SEQ_   197
#define DIM_   192
#define C_     8
#define H_     14
#define W_     14
#define NP_    196
#define TPAD_  208   /* s_t row stride: 13*16, absorbs last-tile overhang */
#define THETA_ 0.7f
#define MT1_   13    /* ceil(196/16) M-tiles */
#define NT2_   12    /* 192/16 N-tiles       */

static __device__ __forceinline__ v8f wmma_f32(v2f a, v2f b, v8f c) {
    return __builtin_amdgcn_wmma_f32_16x16x4_f32(false, a, false, b,
                                                 (short)0, c, false, false);
}

__launch_bounds__(256)
__global__ void sadapter_fused(const float* __restrict__ x,
                               const float* __restrict__ W1,
                               const float* __restrict__ b1,
                               const float* __restrict__ conv1_w,
                               const float* __restrict__ conv1_b,
                               const float* __restrict__ cdc_w,
                               const float* __restrict__ wc1,
                               const float* __restrict__ wc2,
                               const float* __restrict__ hist_w,
                               const float* __restrict__ hist_b,
                               float* __restrict__ out)
{
    __shared__ float s_t  [C_][TPAD_];    // t (channel-major planes, padded)
    __shared__ float s_zs [C_ * NP_];     // z_star, flat (c*196 + p)
    __shared__ float s_e  [C_ * NP_];     // exp(-u^2), flat
    __shared__ float s_zh [C_ * NP_];     // histogram, flat -> GEMM2 raw-reshape reads
    __shared__ float s_W1 [C_ * DIM_];    // W1 staged (c*192 + k)
    __shared__ float s_hw [DIM_ * C_];    // hist_w staged (d*8 + c)
    __shared__ float s_hb [DIM_];
    __shared__ float s_wcen[C_][C_];
    __shared__ float s_w5 [C_][C_][5];
    __shared__ float s_wc1[C_][C_];
    __shared__ float s_wc2[C_][C_];
    __shared__ float s_b1 [C_];
    __shared__ float s_c1b[C_];

    const int tid  = threadIdx.x;
    const int lane = tid & 31;
    const int wave = tid >> 5;
    const int lh   = lane & 15;   // row/column within 16-wide tile
    const int hi   = lane >> 4;   // 0 = lanes 0-15, 1 = lanes 16-31
    const int b    = blockIdx.x;

    // ---- Phase 0: stage small operands in LDS ----
    for (int i = tid; i < C_ * DIM_; i += 256) s_W1[i] = W1[i];
    for (int i = tid; i < DIM_ * C_; i += 256) s_hw[i] = hist_w[i];
    for (int i = tid; i < DIM_;      i += 256) s_hb[i] = hist_b[i];
    if (tid < C_ * C_) {
        const int o = tid >> 3, i = tid & 7;
        const float t0 = cdc_w[(o * C_ + i) * 5 + 0];
        const float t1 = cdc_w[(o * C_ + i) * 5 + 1];
        const float t2 = cdc_w[(o * C_ + i) * 5 + 2];
        const float t3 = cdc_w[(o * C_ + i) * 5 + 3];
        const float t4 = cdc_w[(o * C_ + i) * 5 + 4];
        s_w5[o][i][0] = t0; s_w5[o][i][1] = t1; s_w5[o][i][2] = t2;
        s_w5[o][i][3] = t3; s_w5[o][i][4] = t4;
        s_wcen[o][i] = conv1_w[o * C_ + i] + t2 - THETA_ * (t0 + t1 + t2 + t3 + t4);
        s_wc1[o][i]  = wc1[o * C_ + i];
        s_wc2[o][i]  = wc2[o * C_ + i];
    }
    if (tid < C_) { s_b1[tid] = b1[tid]; s_c1b[tid] = conv1_b[tid]; }
    __syncthreads();

    // ---- Phase A: GEMM1  t[n,c] = sum_d tok[n,d]*W1[c,d], two M-tiles per wave ----
    // A 16x4 f32: lane<16 -> (M=lh, K=k0..k0+1); lane>=16 -> (M=lh, K=k0+2..k0+3)
    // B 4x16 f32: same K split, N=lh;  B[k,n] = W1[n,k]
    // Bounds by ADDRESS CLAMP: garbage lands only in D rows/cols masked at store.
    // Depth-2 software pipeline: loads for step k+8 issue before WMMAs of step k.
    {
        const float* tok = x + (size_t)b * SEQ_ * DIM_ + DIM_;  // skip class token
        const int  mt0  = wave;                       // tiles 0..7  (always valid)
        const int  mt1r = wave + 8;                   // tiles 8..15
        const bool st1  = mt1r < MT1_;
        const int  mt1  = st1 ? mt1r : (MT1_ - 1);
        int r1 = mt1 * 16 + lh; if (r1 >= NP_) r1 = NP_ - 1;          // clamp rows
        const float* a0p = tok + (size_t)(mt0 * 16 + lh) * DIM_ + hi * 2;
        const float* a1p = tok + (size_t)r1 * DIM_ + hi * 2;
        const float* bp  = s_W1 + (size_t)((lh < C_) ? lh : (C_ - 1)) * DIM_ + hi * 2;

        v8f acc0 = {0.f,0.f,0.f,0.f,0.f,0.f,0.f,0.f};
        v8f acc1 = {0.f,0.f,0.f,0.f,0.f,0.f,0.f,0.f};

        v2f bf0 = *(const v2f*)(bp  + 0);
        v2f a00 = *(const v2f*)(a0p + 0);
        v2f a10 = *(const v2f*)(a1p + 0);
        v2f bf1 = *(const v2f*)(bp  + 4);
        v2f a01 = *(const v2f*)(a0p + 4);
        v2f a11 = *(const v2f*)(a1p + 4);
        #pragma unroll 8
        for (int k0 = 0; k0 < DIM_ - 8; k0 += 4) {
            const v2f bfN = *(const v2f*)(bp  + k0 + 8);
            const v2f a0N = *(const v2f*)(a0p + k0 + 8);
            const v2f a1N = *(const v2f*)(a1p + k0 + 8);
            acc0 = wmma_f32(a00, bf0, acc0);
            acc1 = wmma_f32(a10, bf0, acc1);
            bf0 = bf1; a00 = a01; a10 = a11;
            bf1 = bfN; a01 = a0N; a11 = a1N;
        }
        acc0 = wmma_f32(a00, bf0, acc0);
        acc1 = wmma_f32(a10, bf0, acc1);
        acc0 = wmma_f32(a01, bf1, acc0);
        acc1 = wmma_f32(a11, bf1, acc1);

        // D: VGPR r, lane<16 -> (M=r, N=lane); lane>=16 -> (M=8+r, N=lane-16)
        if (lh < C_) {
            const float bias = s_b1[lh];
            #pragma unroll
            for (int r = 0; r < 8; ++r)
                s_t[lh][mt0 * 16 + hi * 8 + r] = acc0[r] + bias;     // p <= 127
            if (st1) {
                #pragma unroll
                for (int r = 0; r < 8; ++r)
                    s_t[lh][mt1r * 16 + hi * 8 + r] = acc1[r] + bias; // overhang -> pad
            }
        }
    }
    __syncthreads();

    // ---- Phase B1: z_star = folded-center 1x1 + cross taps + conv1_b ----
    for (int j = tid; j < C_ * NP_; j += 256) {
        const int o = j / NP_, p = j % NP_;
        const int h = p / W_,  w = p % W_;
        float acc = s_c1b[o];
        #pragma unroll
        for (int i = 0; i < C_; ++i) {
            acc += s_wcen[o][i] * s_t[i][p];
            const float up    = (h > 0)      ? s_t[i][p - W_] : 0.f;
            const float left  = (w > 0)      ? s_t[i][p - 1 ] : 0.f;
            const float right = (w < W_ - 1) ? s_t[i][p + 1 ] : 0.f;
            const float down  = (h < H_ - 1) ? s_t[i][p + W_] : 0.f;
            acc += s_w5[o][i][0] * up + s_w5[o][i][1] * left
                 + s_w5[o][i][3] * right + s_w5[o][i][4] * down;
        }
        s_zs[j] = acc;
    }
    __syncthreads();

    // ---- Phase B2: mu/gamma mixes and e = exp(-(gamma*(z-mu))^2) fused ----
    for (int j = tid; j < C_ * NP_; j += 256) {
        const int o = j / NP_, p = j % NP_;
        float m = 0.f, g = 0.f;
        #pragma unroll
        for (int i = 0; i < C_; ++i) {
            const float z = s_zs[i * NP_ + p];
            m += s_wc1[o][i] * z;
            g += s_wc2[o][i] * z;
        }
        const float u = g * (s_zs[j] - m);
        s_e[j] = __expf(-u * u);
    }
    __syncthreads();

    // ---- Phase C: 3x3 box-sum; zero-padded neighbors contribute exp(0)=1 ----
    for (int j = tid; j < C_ * NP_; j += 256) {
        const int o = j / NP_, p = j % NP_;
        const int h = p / W_,  w = p % W_;
        float sum = 0.f;
        #pragma unroll
        for (int dy = -1; dy <= 1; ++dy)
            #pragma unroll
            for (int dx = -1; dx <= 1; ++dx) {
                const int nh = h + dy, nw = w + dx;
                const bool ib = (nh >= 0) & (nh < H_) & (nw >= 0) & (nw < W_);
                sum += ib ? s_e[o * NP_ + nh * W_ + nw] : 1.0f;
            }
        s_zh[j] = sum * (1.0f / 9.0f);
    }
    __syncthreads();

    // ---- Phase D: GEMM2  y[n,d] = sum_c zh_raw[n,c]*hist_w[d,c] + hist_b[d] ----
    // Raw reshape: zh_raw[n,c] = s_zh[n*8 + c]  (flat (C,H,W) reinterpretation)
    {
        float* yb = out + (size_t)b * SEQ_ * DIM_ + DIM_;
        for (int tile = wave; tile < MT1_ * NT2_; tile += 8) {
            const int mt = tile % MT1_, nt = tile / MT1_;
            const int m0 = mt * 16,     n0 = nt * 16;
            int nc = m0 + lh; if (nc >= NP_) nc = NP_ - 1;   // clamp: masked at store
            const int d = n0 + lh;
            const int kk = hi * 2;
            const v2f a0 = *(const v2f*)(s_zh + nc * C_ + kk);      // ds_load_b64
            const v2f b0 = *(const v2f*)(s_hw + d  * C_ + kk);      // ds_load_b64
            const v2f a1 = *(const v2f*)(s_zh + nc * C_ + kk + 4);
            const v2f b1v = *(const v2f*)(s_hw + d  * C_ + kk + 4);
            v8f acc = {0.f,0.f,0.f,0.f,0.f,0.f,0.f,0.f};
            acc = wmma_f32(a0, b0,  acc);
            acc = wmma_f32(a1, b1v, acc);
            const float hb = s_hb[d];
            #pragma unroll
            for (int r = 0; r < 8; ++r) {
                const int p = m0 + hi * 8 + r;
                if (p < NP_) yb[(size_t)p * DIM_ + d] = acc[r] + hb;
            }
        }
    }

    // ---- Phase E: pass class token through ----
    for (int dd = tid; dd < DIM_; dd += 256)
        out[(size_t)b * SEQ_ * DIM_ + dd] = x[(size_t)b * SEQ_ * DIM_ + dd];
}

extern "C" void kernel_launch(void* const* d_in, const int* in_sizes, int n_in,
                              void* d_out, int out_size, void* d_ws, size_t ws_size,
                              hipStream_t stream) {
    const int batch = in_sizes[0] / (SEQ_ * DIM_);
    sadapter_fused<<<dim3(batch), dim3(256), 0, stream>>>(
        (const float*)d_in[0], (const float*)d_in[1], (const float*)d_in[2],
        (const float*)d_in[3], (const float*)d_in[4], (const float*)d_in[5],
        (const float*)d_in[6], (const float*)d_in[7], (const float*)d_in[8],
        (const float*)d_in[9], (float*)d_out);
}